// SRU_59167469469975
// MI455X (gfx1250) — compile-verified
//
#include <hip/hip_runtime.h>
#include <math.h>

typedef float v2f __attribute__((ext_vector_type(2)));
typedef float v8f __attribute__((ext_vector_type(8)));

#define BSZ   16
#define LSEQ  2048
#define DIN   1024
#define HDIM  512
#define LC    32            // timesteps per chunk (2 WMMA M-tiles)
#define NCH   (LSEQ / LC)   // 64 chunks
#define KB    32            // K rows staged in LDS per stage
#define APAD  36            // sA row stride (floats), conflict-free
#define KBP   34            // sW row stride (floats), conflict-free
#define WAVES 4
#define HT    (WAVES * 16)  // 64 h-channels per block

__global__ __launch_bounds__(128)
void sru_fused_wmma(const float* __restrict__ x,   // (B, L, DIN)
                    const float* __restrict__ W,   // (DIN, 2, H, 3)
                    const float* __restrict__ bf,  // (2, H)
                    const float* __restrict__ br,  // (2, H)
                    float* __restrict__ out)       // h (B,L,2H) ++ cT (2H,B)
{
    __shared__ float sA[LC * APAD];              // x tile: 32 x 32 (padded)
    __shared__ float sW[192 * KBP];              // W tile, transposed: [col][k]
    __shared__ float sU[WAVES][3][LC * 17];      // per-wave U tiles (xtil,f,r)

    const int tid  = threadIdx.x;
    const int wave = tid >> 5;
    const int lane = tid & 31;
    const int ln   = lane & 15;       // M (A), N (B/D) sub-index
    const int hh   = lane >> 4;       // half-wave select

    // grid = B * 2 * (H/HT)
    int blk = blockIdx.x;
    const int htile = blk % (HDIM / HT);  blk /= (HDIM / HT);
    const int dir   = blk % 2;            blk /= 2;
    const int b     = blk;
    const int h     = htile * HT + wave * 16 + ln;   // scan lane's h channel
    const int cb    = (wave * 16 + ln) * 3;          // lane's base column in sW

    const float bfv = bf[dir * HDIM + h];
    const float brv = br[dir * HDIM + h];

    const float* __restrict__ xb   = x   + (size_t)b * LSEQ * DIN;
    float*       __restrict__ outb = out + (size_t)b * LSEQ * (2 * HDIM);
    // block's W slice: for K row i, cols (h*3+g) are contiguous 192 floats
    const float* __restrict__ Wblk = W + (size_t)dir * (HDIM * 3) + (size_t)htile * (HT * 3);

    float c = 0.0f;   // recurrent cell state, carried across all chunks

    for (int ci = 0; ci < NCH; ++ci) {
        const int cc = dir ? (NCH - 1 - ci) : ci;   // backward dir walks L in reverse
        const int l0 = cc * LC;

        // prefetch next chunk's x rows into GL2
        if (ci + 1 < NCH) {
            const int cn = dir ? (cc - 1) : (cc + 1);
            __builtin_prefetch(xb + (size_t)(cn * LC + (tid & 31)) * DIN + (tid >> 5) * 256, 0, 1);
        }

        v8f acc0a = {}, acc0b = {};   // gate 0 (xtil), M-tiles 0/1
        v8f acc1a = {}, acc1b = {};   // gate 1 (f)
        v8f acc2a = {}, acc2b = {};   // gate 2 (r)

        for (int kb = 0; kb < DIN; kb += KB) {
            __syncthreads();
            // --- cooperative coalesced load: x tile 32x32 ---
            #pragma unroll
            for (int i = 0; i < (LC * KB) / 128; ++i) {   // 8
                int idx = tid + i * 128;
                int m = idx >> 5;            // row (timestep)
                int j = idx & (KB - 1);      // col (k)
                sA[m * APAD + j] = xb[(size_t)(l0 + m) * DIN + kb + j];
            }
            // --- cooperative coalesced load: W tile, transposed into [col][k] ---
            const float* Wg = Wblk + (size_t)kb * (2 * HDIM * 3);
            for (int kr0 = 0; kr0 < KB; kr0 += 2) {       // 2 rows (384 floats) per 3 passes
                #pragma unroll
                for (int p = 0; p < 3; ++p) {
                    int idx = p * 128 + tid;              // 0..383
                    int kr  = kr0 + (idx >= 192 ? 1 : 0);
                    int col = idx - (idx >= 192 ? 192 : 0);
                    sW[col * KBP + kr] = Wg[(size_t)kr * (2 * HDIM * 3) + col];
                }
            }
            __syncthreads();

            // --- WMMA over staged K chunk ---
            #pragma unroll
            for (int kk = 0; kk < KB; kk += 4) {
                const int ko = kk + 2 * hh;   // this lane's K pair
                v2f a0 = *(const v2f*)&sA[ln * APAD + ko];          // M-tile 0 (rows 0-15)
                v2f a1 = *(const v2f*)&sA[(16 + ln) * APAD + ko];   // M-tile 1 (rows 16-31)
                v2f b0 = *(const v2f*)&sW[(cb + 0) * KBP + ko];     // xtil
                v2f b1 = *(const v2f*)&sW[(cb + 1) * KBP + ko];     // f
                v2f b2 = *(const v2f*)&sW[(cb + 2) * KBP + ko];     // r

                acc0a = __builtin_amdgcn_wmma_f32_16x16x4_f32(false, a0, false, b0, (short)0, acc0a, false, false);
                acc0b = __builtin_amdgcn_wmma_f32_16x16x4_f32(false, a1, false, b0, (short)0, acc0b, false, false);
                acc1a = __builtin_amdgcn_wmma_f32_16x16x4_f32(false, a0, false, b1, (short)0, acc1a, false, false);
                acc1b = __builtin_amdgcn_wmma_f32_16x16x4_f32(false, a1, false, b1, (short)0, acc1b, false, false);
                acc2a = __builtin_amdgcn_wmma_f32_16x16x4_f32(false, a0, false, b2, (short)0, acc2a, false, false);
                acc2b = __builtin_amdgcn_wmma_f32_16x16x4_f32(false, a1, false, b2, (short)0, acc2b, false, false);
            }
        }

        // dump U tiles (D layout: VGPR v -> M = v + 8*hh, lane ln -> N)
        float* su0 = &sU[wave][0][0];
        float* su1 = &sU[wave][1][0];
        float* su2 = &sU[wave][2][0];
        #pragma unroll
        for (int v = 0; v < 8; ++v) {
            const int M = v + 8 * hh;
            su0[M * 17 + ln] = acc0a[v];  su0[(16 + M) * 17 + ln] = acc0b[v];
            su1[M * 17 + ln] = acc1a[v];  su1[(16 + M) * 17 + ln] = acc1b[v];
            su2[M * 17 + ln] = acc2a[v];  su2[(16 + M) * 17 + ln] = acc2b[v];
        }
        __syncthreads();

        // sequential scan: lanes 0-15, each owns one h channel, 32 timesteps
        if (lane < 16) {
            #pragma unroll 1
            for (int m = 0; m < LC; ++m) {
                const int mm = dir ? (LC - 1 - m) : m;
                const int l  = l0 + mm;
                const float xt = su0[mm * 17 + ln];
                const float fp = su1[mm * 17 + ln] + bfv;
                const float rp = su2[mm * 17 + ln] + brv;
                const float f  = 1.0f / (1.0f + __expf(-fp));
                const float r  = 1.0f / (1.0f + __expf(-rp));
                c = f * c + (1.0f - f) * xt;
                const float xres = xb[(size_t)l * DIN + dir * HDIM + h];
                const float hv   = r * tanhf(c) + (1.0f - r) * xres;
                outb[(size_t)l * (2 * HDIM) + dir * HDIM + h] = hv;
            }
        }
        __syncthreads();
    }

    // final cell state: out tail is c.T with shape (2H, B)
    if (lane < 16) {
        out[(size_t)BSZ * LSEQ * (2 * HDIM) + (size_t)(dir * HDIM + h) * BSZ + b] = c;
    }
}

extern "C" void kernel_launch(void* const* d_in, const int* in_sizes, int n_in,
                              void* d_out, int out_size, void* d_ws, size_t ws_size,
                              hipStream_t stream) {
    (void)in_sizes; (void)n_in; (void)out_size; (void)d_ws; (void)ws_size;
    const float* x  = (const float*)d_in[0];
    const float* W  = (const float*)d_in[1];
    const float* bf = (const float*)d_in[2];
    const float* br = (const float*)d_in[3];
    float* out = (float*)d_out;

    dim3 grid(BSZ * 2 * (HDIM / HT));   // 256 blocks
    dim3 block(WAVES * 32);             // 128 threads = 4 waves
    sru_fused_wmma<<<grid, block, 0, stream>>>(x, W, bf, br, out);
}